// BAGDnet_66657892434512
// MI455X (gfx1250) — compile-verified
//
#include <hip/hip_runtime.h>

// ---- problem constants (pinhole intrinsics from the reference) ----
#define K_FX 320.0f
#define K_FY 320.0f
#define K_CX 320.0f
#define K_CY 240.0f

#define MAX_KF     2000
#define KF_FLOATS  (MAX_KF * 16)     // 32000 floats = 128000 bytes of LDS
#define BLOCK      256
#define GRID       1024

#define AS1 __attribute__((address_space(1)))
#define AS3 __attribute__((address_space(3)))

#if defined(__has_builtin)
#if __has_builtin(__builtin_amdgcn_global_load_async_to_lds_b128)
#define HAVE_ASYNC_LDS 1
#endif
#if __has_builtin(__builtin_amdgcn_s_wait_asynccnt)
#define HAVE_WAIT_ASYNC 1
#endif
#endif

typedef __attribute__((ext_vector_type(2))) float v2f;
typedef __attribute__((ext_vector_type(4))) float v4f;
typedef __attribute__((ext_vector_type(4))) int   v4i;

// searchsorted-left: first i with t[i] >= v (tables are sorted unique)
__device__ __forceinline__ int lower_bound(const int* __restrict__ t, int n, int v) {
    int lo = 0, hi = n;
    while (lo < hi) {
        int mid = (lo + hi) >> 1;
        if (t[mid] < v) lo = mid + 1; else hi = mid;
    }
    return lo;
}

__global__ __launch_bounds__(BLOCK) void bagd_project_kernel(
    const float* __restrict__ tMP,     // [N_MP,3]
    const float* __restrict__ tKF,     // [N_KF,4,4]
    const int*   __restrict__ idxKF,   // [N_KF] sorted unique
    const int*   __restrict__ idxMP,   // [N_MP] sorted unique
    const int*   __restrict__ meas_kf, // [M]
    const int*   __restrict__ meas_mp, // [M]
    float*       __restrict__ out,     // [M,2]
    int n_kf, int n_mp, int M)
{
    // Pose table (128000 B) + KF id table (8000 B) staged in LDS.
    __shared__ float ldsKF[KF_FLOATS];
    __shared__ int   ldsIdxKF[MAX_KF];

    const int tid    = threadIdx.x;
    const int nf4    = (n_kf * 16) >> 2;   // float4 chunks of pose table (<= 8000)
    const int nid4   = (n_kf + 3) >> 2;    // int4  chunks of id table   (<= 500)

    // ---- stage tKF and idxKF -> LDS via async DMA (ASYNCcnt path) ----
#if defined(HAVE_ASYNC_LDS)
    {
        const v4f* g4 = (const v4f*)tKF;
        v4f*       l4 = (v4f*)ldsKF;
        for (int i = tid; i < nf4; i += BLOCK) {
            __builtin_amdgcn_global_load_async_to_lds_b128(
                (AS1 v4i*)(g4 + i), (AS3 v4i*)(l4 + i), 0, 0);
        }
        const v4i* gi4 = (const v4i*)idxKF;
        v4i*       li4 = (v4i*)ldsIdxKF;
        for (int i = tid; i < nid4; i += BLOCK) {
            __builtin_amdgcn_global_load_async_to_lds_b128(
                (AS1 v4i*)(gi4 + i), (AS3 v4i*)(li4 + i), 0, 0);
        }
#if defined(HAVE_WAIT_ASYNC)
        __builtin_amdgcn_s_wait_asynccnt(0);
#else
        asm volatile("s_wait_asynccnt 0" ::: "memory");
#endif
    }
#else
    {   // fallback: copy through VGPRs
        const v4f* g4 = (const v4f*)tKF;
        v4f*       l4 = (v4f*)ldsKF;
        for (int i = tid; i < nf4; i += BLOCK) l4[i] = g4[i];
        const v4i* gi4 = (const v4i*)idxKF;
        v4i*       li4 = (v4i*)ldsIdxKF;
        for (int i = tid; i < nid4; i += BLOCK) li4[i] = gi4[i];
    }
#endif
    __syncthreads();

    // ---- grid-stride over measurements ----
    const int stride = GRID * BLOCK;
    for (int m = blockIdx.x * BLOCK + tid; m < M; m += stride) {
        // streaming, read-once: keep out of caches (NT hint)
        const int kf_id = __builtin_nontemporal_load(meas_kf + m);
        const int mp_id = __builtin_nontemporal_load(meas_mp + m);

        // faithful searchsorted: KF table in LDS, MP table L2-hot
        const int ikf = lower_bound(ldsIdxKF, n_kf, kf_id);
        const int imp = lower_bound(idxMP, n_mp, mp_id);

        // gather map point (tMP is 2.4 MB -> L2 resident)
        const float* p = tMP + (size_t)imp * 3;
        const float x = p[0];
        const float y = p[1];
        const float z = p[2];

        // gather pose rows 0..2 from LDS (3x ds_load_b128)
        const float* T = ldsKF + ikf * 16;
        const float tx = fmaf(T[0], x, fmaf(T[1], y, fmaf(T[2],  z, T[3])));
        const float ty = fmaf(T[4], x, fmaf(T[5], y, fmaf(T[6],  z, T[7])));
        const float tz = fmaf(T[8], x, fmaf(T[9], y, fmaf(T[10], z, T[11])));

        const float rz = 1.0f / tz;
        v2f uv;
        uv.x = (tx * rz) * K_FX + K_CX;
        uv.y = (ty * rz) * K_FY + K_CY;

        // streaming, write-once output
        __builtin_nontemporal_store(uv, (v2f*)out + m);
    }
}

extern "C" void kernel_launch(void* const* d_in, const int* in_sizes, int n_in,
                              void* d_out, int out_size, void* d_ws, size_t ws_size,
                              hipStream_t stream) {
    // setup_inputs() order: tMP, tKF, idxKF, idxMP, meas_kf, meas_mp
    const float* tMP     = (const float*)d_in[0];
    const float* tKF     = (const float*)d_in[1];
    const int*   idxKF   = (const int*)d_in[2];
    const int*   idxMP   = (const int*)d_in[3];
    const int*   meas_kf = (const int*)d_in[4];
    const int*   meas_mp = (const int*)d_in[5];
    float*       out     = (float*)d_out;

    const int n_kf = in_sizes[2];
    const int n_mp = in_sizes[3];
    const int M    = in_sizes[4];

    bagd_project_kernel<<<GRID, BLOCK, 0, stream>>>(
        tMP, tKF, idxKF, idxMP, meas_kf, meas_mp, out, n_kf, n_mp, M);
}